// ResidualGCN_4063039062434
// MI455X (gfx1250) — compile-verified
//
#include <hip/hip_runtime.h>
#include <hip/hip_bf16.h>

#define NNODES 100000
#define NEDGES 800000
#define INDIM  128
#define HIDDIM 256
#define BN_EPS 1e-5f

typedef __attribute__((ext_vector_type(2))) float v2f;
typedef __attribute__((ext_vector_type(4))) float v4f;
typedef __attribute__((ext_vector_type(8))) float v8f;

// ---------------------------------------------------------------------------
// WMMA fp32 GEMM: C[n x HIDDIM] = A[n x K] * B[K x HIDDIM]
// Workgroup: 256 threads = 8 waves. Each wave computes a 16x64 output tile
// (4x v_wmma_f32_16x16x4_f32 accumulators). K tiled by 64 through LDS.
// B is staged K-pair interleaved: Bs[(k>>1)*128 + n*2 + (k&1)], so each
// lane's fragment (B[k][n], B[k+1][n]) is one contiguous ds_load_b64 into an
// aligned VGPR pair -- no packing movs in the WMMA loop.
// ---------------------------------------------------------------------------
__global__ __launch_bounds__(256) void gemm_wmma_f32(
    const float* __restrict__ A, const float* __restrict__ B,
    float* __restrict__ C, int n, int K)
{
    __shared__ float Bs[64 * 64];        // 16 KB: B chunk, K-pair interleaved
    __shared__ float As[8][16 * 64];     // 32 KB: per-wave A chunk, [m][k] stride 64

    const int tid  = threadIdx.x;
    const int wave = tid >> 5;
    const int lane = tid & 31;
    const int l16  = lane & 15;
    const int lh   = lane >> 4;                    // 0: K pair {0,1}, 1: {2,3}
    const int rowTile = blockIdx.x * 128 + wave * 16;
    const int colBase = blockIdx.y * 64;
    const bool valid  = (rowTile < n);             // wave-uniform (N % 16 == 0)

    v8f acc0 = {}, acc1 = {}, acc2 = {}, acc3 = {};

    for (int kc = 0; kc < K; kc += 64) {
        // Stage B chunk (64K x 64N) -> LDS, K-pair interleaved layout
        for (int i = tid * 4; i < 64 * 64; i += 256 * 4) {
            const int kr = i >> 6, cl = i & 63;
            const v4f bv = *(const v4f*)&B[(size_t)(kc + kr) * HIDDIM + colBase + cl];
            float* bs = &Bs[(kr >> 1) * 128 + cl * 2 + (kr & 1)];
            bs[0] = bv[0];
            bs[2] = bv[1];
            bs[4] = bv[2];
            bs[6] = bv[3];
        }
        // Stage this wave's A chunk (16M x 64K) -> LDS, coalesced float4
        if (valid) {
            for (int i = lane * 4; i < 16 * 64; i += 32 * 4) {
                const int r = i >> 6, cl = i & 63;
                *(v4f*)&As[wave][i] = *(const v4f*)&A[(size_t)(rowTile + r) * K + kc + cl];
            }
        }
        __syncthreads();

        if (valid) {
            #pragma unroll
            for (int kk = 0; kk < 64; kk += 4) {
                // A frag: lane l (l<16): A[M=l][kk+0..1]; lane l>=16: A[M=l-16][kk+2..3]
                const v2f a = *(const v2f*)&As[wave][l16 * 64 + kk + 2 * lh];
                // B frags: kp = (kk + 2*lh)>>1 selects the interleaved K-pair row
                const float* brow = &Bs[((kk + 2 * lh) >> 1) * 128 + l16 * 2];
                const v2f b0 = *(const v2f*)(brow +  0);
                const v2f b1 = *(const v2f*)(brow + 32);
                const v2f b2 = *(const v2f*)(brow + 64);
                const v2f b3 = *(const v2f*)(brow + 96);
                acc0 = __builtin_amdgcn_wmma_f32_16x16x4_f32(false, a, false, b0, (short)0, acc0, false, false);
                acc1 = __builtin_amdgcn_wmma_f32_16x16x4_f32(false, a, false, b1, (short)0, acc1, false, false);
                acc2 = __builtin_amdgcn_wmma_f32_16x16x4_f32(false, a, false, b2, (short)0, acc2, false, false);
                acc3 = __builtin_amdgcn_wmma_f32_16x16x4_f32(false, a, false, b3, (short)0, acc3, false, false);
            }
        }
        __syncthreads();
    }

    if (valid) {
        // D layout: lanes 0-15: VGPR r -> M=r; lanes 16-31: VGPR r -> M=8+r; N=l16
        const int row0 = rowTile + 8 * lh;
        #pragma unroll
        for (int r = 0; r < 8; ++r) {
            float* crow = C + (size_t)(row0 + r) * HIDDIM + colBase + l16;
            crow[ 0] = acc0[r];
            crow[16] = acc1[r];
            crow[32] = acc2[r];
            crow[48] = acc3[r];
        }
    }
}

// ---------------------------------------------------------------------------
// Helpers
// ---------------------------------------------------------------------------
__global__ void k_fill(float* __restrict__ p, float v, int nElem) {
    const int i = blockIdx.x * blockDim.x + threadIdx.x;
    if (i < nElem) p[i] = v;
}

__global__ void k_deg_count(const int* __restrict__ dst, float* __restrict__ deg) {
    const int e = blockIdx.x * blockDim.x + threadIdx.x;
    if (e < NEDGES) atomicAdd(&deg[dst[e]], 1.0f);
}

__global__ void k_edge_norm(const int* __restrict__ src, const int* __restrict__ dst,
                            const float* __restrict__ deg, float* __restrict__ nrm) {
    const int e = blockIdx.x * blockDim.x + threadIdx.x;
    if (e < NEDGES) nrm[e] = rsqrtf(deg[src[e]]) * rsqrtf(deg[dst[e]]);
}

// 64 threads per edge, float4 per thread: gather h[src], atomic-add into agg[dst].
// h (102 MB) is L2-resident on MI455X (192 MB L2), so this stays on-chip.
__global__ __launch_bounds__(256) void k_scatter(
    const int* __restrict__ src, const int* __restrict__ dst,
    const float* __restrict__ nrm,
    const float* __restrict__ H, float* __restrict__ AGG)
{
    const int t = blockIdx.x * blockDim.x + threadIdx.x;   // exactly E*64 threads
    const int e = t >> 6;
    const int g = (t & 63) << 2;
    const int s = src[e], d = dst[e];
    const float w = nrm[e];
    const v4f v = *(const v4f*)&H[(size_t)s * HIDDIM + g];
    float* o = &AGG[(size_t)d * HIDDIM + g];
    atomicAdd(o + 0, v[0] * w);
    atomicAdd(o + 1, v[1] * w);
    atomicAdd(o + 2, v[2] * w);
    atomicAdd(o + 3, v[3] * w);
}

// Fused: z = agg + h*(1/deg) + b   (written back to AGG) + per-column BN partials.
// One block = 256 cols x 64 rows; per-column partial sums then 2 atomics/col.
__global__ __launch_bounds__(256) void k_stats(
    float* __restrict__ AGG, const float* __restrict__ H,
    const float* __restrict__ deg, const float* __restrict__ bias,
    float* __restrict__ colsum, float* __restrict__ colsq)
{
    const int c = threadIdx.x;
    const int row0 = blockIdx.x * 64;
    const int rend = min(row0 + 64, NNODES);
    const float b = bias[c];
    float s = 0.0f, q = 0.0f;
    for (int r = row0; r < rend; ++r) {
        const float dinv = 1.0f / deg[r];
        const size_t off = (size_t)r * HIDDIM + c;
        const float z = AGG[off] + H[off] * dinv + b;
        AGG[off] = z;
        s += z;
        q += z * z;
    }
    atomicAdd(&colsum[c], s);
    atomicAdd(&colsq[c], q);
}

__global__ void k_bn_final(const float* __restrict__ colsum, const float* __restrict__ colsq,
                           const float* __restrict__ g, const float* __restrict__ be,
                           float* __restrict__ scale, float* __restrict__ shift)
{
    const int c = threadIdx.x;   // 256 threads
    const float mean = colsum[c] * (1.0f / (float)NNODES);
    const float var  = colsq[c] * (1.0f / (float)NNODES) - mean * mean;
    const float rs   = rsqrtf(var + BN_EPS);
    const float sc   = rs * g[c];
    scale[c] = sc;
    shift[c] = be[c] - mean * sc;
}

// out = pre + relu(z*scale + shift)
__global__ __launch_bounds__(256) void k_apply(
    const float* __restrict__ AGG, const float* __restrict__ PRE,
    const float* __restrict__ scale, const float* __restrict__ shift,
    float* __restrict__ OUT)
{
    const size_t i = (size_t)blockIdx.x * blockDim.x + threadIdx.x;
    const int c = (int)(i & (HIDDIM - 1));
    const float z = AGG[i] * scale[c] + shift[c];
    OUT[i] = PRE[i] + fmaxf(z, 0.0f);
}

// ---------------------------------------------------------------------------
// Orchestration
// ---------------------------------------------------------------------------
extern "C" void kernel_launch(void* const* d_in, const int* in_sizes, int n_in,
                              void* d_out, int out_size, void* d_ws, size_t ws_size,
                              hipStream_t stream) {
    (void)in_sizes; (void)n_in; (void)out_size; (void)ws_size;

    // setup_inputs order: edge_index, x, W1, b1, g1, be1, W2, b2, g2, be2, Wres
    const int*   edge = (const int*)d_in[0];   // (2,E) int32 (JAX default x64-off)
    const float* x    = (const float*)d_in[1];
    const float* W1   = (const float*)d_in[2];
    const float* b1   = (const float*)d_in[3];
    const float* g1   = (const float*)d_in[4];
    const float* be1  = (const float*)d_in[5];
    const float* W2   = (const float*)d_in[6];
    const float* b2   = (const float*)d_in[7];
    const float* g2   = (const float*)d_in[8];
    const float* be2  = (const float*)d_in[9];
    const float* Wres = (const float*)d_in[10];
    const int* src = edge;
    const int* dst = edge + NEDGES;

    float* out = (float*)d_out;

    // Workspace layout (floats)
    float* ws     = (float*)d_ws;
    float* deg    = ws;                                  // N
    float* nrm    = deg + NNODES;                        // E
    float* Hbuf   = nrm + NEDGES;                        // N*HID
    float* PRE    = Hbuf + (size_t)NNODES * HIDDIM;      // N*HID
    float* AGG    = PRE + (size_t)NNODES * HIDDIM;       // N*HID
    float* colsum = AGG + (size_t)NNODES * HIDDIM;       // HID
    float* colsq  = colsum + HIDDIM;                     // HID
    float* scale  = colsq + HIDDIM;                      // HID
    float* shift  = scale + HIDDIM;                      // HID

    const int NH       = NNODES * HIDDIM;                // 25,600,000
    const int nhBlocks = NH / 256;                       // 100,000
    const dim3 gemmGrid((NNODES + 127) / 128, HIDDIM / 64);  // (782, 4)
    const int edgeBlocks    = (NEDGES + 255) / 256;          // 3125
    const int scatterBlocks = NEDGES * 64 / 256;             // 200,000
    const int statsBlocks   = (NNODES + 63) / 64;            // 1563

    // --- degrees & symmetric norm (shared by both conv layers) ---
    k_fill<<<(NNODES + 255) / 256, 256, 0, stream>>>(deg, 1.0f, NNODES); // +self loop
    k_deg_count<<<edgeBlocks, 256, 0, stream>>>(dst, deg);
    k_edge_norm<<<edgeBlocks, 256, 0, stream>>>(src, dst, deg, nrm);

    // --- layer 1: PRE = x@Wres, H = x@W1 ---
    gemm_wmma_f32<<<gemmGrid, 256, 0, stream>>>(x, Wres, PRE, NNODES, INDIM);
    gemm_wmma_f32<<<gemmGrid, 256, 0, stream>>>(x, W1, Hbuf, NNODES, INDIM);

    k_fill<<<nhBlocks, 256, 0, stream>>>(AGG, 0.0f, NH);
    k_fill<<<2, 256, 0, stream>>>(colsum, 0.0f, 2 * HIDDIM);
    k_scatter<<<scatterBlocks, 256, 0, stream>>>(src, dst, nrm, Hbuf, AGG);
    k_stats<<<statsBlocks, 256, 0, stream>>>(AGG, Hbuf, deg, b1, colsum, colsq);
    k_bn_final<<<1, 256, 0, stream>>>(colsum, colsq, g1, be1, scale, shift);
    // PRE <- PRE + relu(bn(z))  : layer-1 output, input & residual for layer 2
    k_apply<<<nhBlocks, 256, 0, stream>>>(AGG, PRE, scale, shift, PRE);

    // --- layer 2: H = PRE@W2 ---
    gemm_wmma_f32<<<gemmGrid, 256, 0, stream>>>(PRE, W2, Hbuf, NNODES, HIDDIM);

    k_fill<<<nhBlocks, 256, 0, stream>>>(AGG, 0.0f, NH);
    k_fill<<<2, 256, 0, stream>>>(colsum, 0.0f, 2 * HIDDIM);
    k_scatter<<<scatterBlocks, 256, 0, stream>>>(src, dst, nrm, Hbuf, AGG);
    k_stats<<<statsBlocks, 256, 0, stream>>>(AGG, Hbuf, deg, b2, colsum, colsq);
    k_bn_final<<<1, 256, 0, stream>>>(colsum, colsq, g2, be2, scale, shift);
    k_apply<<<nhBlocks, 256, 0, stream>>>(AGG, PRE, scale, shift, out);
}